// MoEGate_87600152969589
// MI455X (gfx1250) — compile-verified
//
#include <hip/hip_runtime.h>
#include <hip/hip_bf16.h>
#include <math.h>

typedef __attribute__((ext_vector_type(2))) float v2f;
typedef __attribute__((ext_vector_type(8))) float v8f;

#define NUM_EXPERTS 32
#define TOP_K 6
#define DIM 2048
#define SEQ 4096
#define NBATCH 4
#define NTOK 16384      // NBATCH * SEQ
#define WAVES 4
#define KSLICE (DIM / WAVES)   // 512

// ---------------------------------------------------------------------------
// Zero the per-(batch,expert) count accumulators each call (graph-safe).
// ---------------------------------------------------------------------------
__global__ void moe_zero_counts(float* __restrict__ ws) {
  int t = blockIdx.x * blockDim.x + threadIdx.x;
  if (t < NBATCH * NUM_EXPERTS) ws[t] = 0.0f;
}

// ---------------------------------------------------------------------------
// Main kernel: one block = 16 tokens. 4 waves split K (512 each).
// WMMA f32 16x16x4: A = x tile (16 tokens x 4 K), B = W^T tile (4 K x 16 experts).
// Two N-tiles cover the 32 experts.
// ---------------------------------------------------------------------------
__global__ __launch_bounds__(128) void moe_gate_kernel(
    const float* __restrict__ x, const float* __restrict__ W,
    float* __restrict__ outIdx, float* __restrict__ outW,
    float* __restrict__ cntWs, float* __restrict__ blkSum) {
  __shared__ float partial[WAVES][16][NUM_EXPERTS];
  __shared__ float logitsS[16][NUM_EXPERTS];
  __shared__ float scoresS[16][NUM_EXPERTS + 1];  // +1 pad vs bank conflicts
  __shared__ float cntS[NUM_EXPERTS];

  const int lane = threadIdx.x & 31;
  const int wv   = threadIdx.x >> 5;
  const int row0 = blockIdx.x * 16;

  // ISA 16x4 f32 A layout: lanes 0-15 hold M=lane, K = {0,1}; lanes 16-31 hold
  // M=lane-16, K = {2,3}.  B 4x16 mirrors with N in lanes.
  const int m  = lane & 15;
  const int kp = (lane >> 4) << 1;  // 0 or 2

  const float* xp  = x + (size_t)(row0 + m) * DIM + kp;
  const float* wp0 = W + (size_t)m * DIM + kp;           // experts 0..15
  const float* wp1 = wp0 + (size_t)16 * DIM;             // experts 16..31

  v8f c0 = {};
  v8f c1 = {};
  const int kbeg = wv * KSLICE;
  #pragma unroll 4
  for (int kk = kbeg; kk < kbeg + KSLICE; kk += 4) {
    v2f a  = *(const v2f*)(xp  + kk);
    v2f b0 = *(const v2f*)(wp0 + kk);
    v2f b1 = *(const v2f*)(wp1 + kk);
    // 8 args: (neg_a, A, neg_b, B, c_mod, C, reuse_a, reuse_b)
    c0 = __builtin_amdgcn_wmma_f32_16x16x4_f32(false, a, false, b0, (short)0,
                                               c0, false, false);
    c1 = __builtin_amdgcn_wmma_f32_16x16x4_f32(false, a, false, b1, (short)0,
                                               c1, false, false);
  }

  // C/D layout: VGPR r, lane L -> row M = r + 8*(L>>4), col N = L&15.
  const int mh = (lane >> 4) * 8;
  #pragma unroll
  for (int r = 0; r < 8; ++r) {
    partial[wv][mh + r][m]      = c0[r];
    partial[wv][mh + r][16 + m] = c1[r];
  }
  if (threadIdx.x < NUM_EXPERTS) cntS[threadIdx.x] = 0.0f;
  __syncthreads();

  // Cross-wave split-K reduction: 512 elems over 128 threads.
  for (int i = threadIdx.x; i < 16 * NUM_EXPERTS; i += 128) {
    int mm = i >> 5, e = i & 31;
    logitsS[mm][e] = partial[0][mm][e] + partial[1][mm][e] +
                     partial[2][mm][e] + partial[3][mm][e];
  }
  __syncthreads();

  // Softmax + top-6 per token: lanes 0..15 of wave 0 each own one token.
  if (threadIdx.x < 16) {
    const int mm = threadIdx.x;
    float l[NUM_EXPERTS];
    float mx = -INFINITY;
    #pragma unroll
    for (int e = 0; e < NUM_EXPERTS; ++e) {
      l[e] = logitsS[mm][e];
      mx = fmaxf(mx, l[e]);
    }
    float s = 0.0f;
    #pragma unroll
    for (int e = 0; e < NUM_EXPERTS; ++e) {
      l[e] = __expf(l[e] - mx);
      s += l[e];
    }
    const float inv = 1.0f / s;
    #pragma unroll
    for (int e = 0; e < NUM_EXPERTS; ++e) {
      l[e] *= inv;
      scoresS[mm][e] = l[e];
    }
    const int tok = row0 + mm;
    unsigned taken = 0u;
    for (int k = 0; k < TOP_K; ++k) {
      float best = -1.0f;
      int bi = 0;
      #pragma unroll
      for (int e = 0; e < NUM_EXPERTS; ++e) {
        // strict '>' keeps lowest index on ties, matching lax.top_k
        if (!((taken >> e) & 1u) && l[e] > best) { best = l[e]; bi = e; }
      }
      taken |= (1u << bi);
      outIdx[tok * TOP_K + k] = (float)bi;
      outW[tok * TOP_K + k]   = best;
      atomicAdd(&cntS[bi], 1.0f);  // integer-valued fp32 adds: exact
    }
  }
  __syncthreads();

  // Per-expert score column-sum for this tile -> deterministic per-block slot;
  // expert counts -> global atomics (exact integer fp32 sums).
  if (threadIdx.x < NUM_EXPERTS) {
    const int e = threadIdx.x;
    float s = 0.0f;
    #pragma unroll
    for (int mm = 0; mm < 16; ++mm) s += scoresS[mm][e];
    blkSum[(size_t)blockIdx.x * NUM_EXPERTS + e] = s;
    const int b = row0 / SEQ;
    atomicAdd(&cntWs[b * NUM_EXPERTS + e], cntS[e]);
  }
}

// ---------------------------------------------------------------------------
// Finalize: deterministic fixed-order reduction of per-block score sums,
// combine with counts -> aux loss scalar.
// ---------------------------------------------------------------------------
__global__ __launch_bounds__(128) void moe_finalize(
    const float* __restrict__ cntWs, const float* __restrict__ blkSum,
    float* __restrict__ auxOut) {
  __shared__ float red[NBATCH * NUM_EXPERTS];
  const int t = threadIdx.x;  // 128 threads, one per (batch, expert)
  const int b = t >> 5;
  const int e = t & 31;
  float ssum = 0.0f;
  const int blocksPerBatch = SEQ / 16;  // 256
  for (int blk = b * blocksPerBatch; blk < (b + 1) * blocksPerBatch; ++blk)
    ssum += blkSum[(size_t)blk * NUM_EXPERTS + e];
  const float ce = cntWs[t] * ((float)NUM_EXPERTS / (float)(SEQ * TOP_K));
  const float ms = ssum / (float)SEQ;
  red[t] = ce * ms;
  __syncthreads();
  for (int s = 64; s > 0; s >>= 1) {
    if (t < s) red[t] += red[t + s];
    __syncthreads();
  }
  if (t == 0) auxOut[0] = 0.001f * (red[0] / (float)NBATCH);
}

// ---------------------------------------------------------------------------
extern "C" void kernel_launch(void* const* d_in, const int* in_sizes, int n_in,
                              void* d_out, int out_size, void* d_ws,
                              size_t ws_size, hipStream_t stream) {
  const float* x = (const float*)d_in[0];  // [4, 4096, 2048] f32
  const float* W = (const float*)d_in[1];  // [32, 2048] f32

  float* out    = (float*)d_out;
  float* outIdx = out;                            // [T*6] indices (as float)
  float* outW   = out + (size_t)NTOK * TOP_K;     // [T*6] weights
  float* aux    = out + (size_t)2 * NTOK * TOP_K; // [1]

  // Workspace: [128] counts | [1024*32] per-block score sums (~128.5 KB).
  float* cntWs  = (float*)d_ws;
  float* blkSum = cntWs + NBATCH * NUM_EXPERTS;

  moe_zero_counts<<<1, 128, 0, stream>>>(cntWs);
  moe_gate_kernel<<<NTOK / 16, 128, 0, stream>>>(x, W, outIdx, outW, cntWs,
                                                 blkSum);
  moe_finalize<<<1, 128, 0, stream>>>(cntWs, blkSum, aux);
}